// StockPredictor_22033182228781
// MI455X (gfx1250) — compile-verified
//
#include <hip/hip_runtime.h>
#include <hip/hip_bf16.h>
#include <math.h>

// ---------------------------------------------------------------------------
// Sizes (fixed by the reference)
// ---------------------------------------------------------------------------
#define BATCH    16
#define SEQ      4096
#define INDIM    64
#define HID      512
#define DI       768            // DIM_INNER
#define NTOK     (BATCH * SEQ)  // 65536 tokens
#define NCH      (BATCH * DI)   // 12288 scan channels
#define NCHUNK   16
#define CLEN     (SEQ / NCHUNK) // 256
#define NDG      (DI / 32)      // 24 d-groups of one wave each

typedef __attribute__((ext_vector_type(2))) float v2f;
typedef __attribute__((ext_vector_type(8))) float v8f;

// ---------------------------------------------------------------------------
// Math helpers (match reference semantics in fp32)
// ---------------------------------------------------------------------------
__device__ __forceinline__ float softplusf(float x) {
    // log(1 + e^x), numerically stable
    return fmaxf(x, 0.0f) + log1pf(expf(-fabsf(x)));
}
__device__ __forceinline__ float log_gf(float h) {
    // x>=0 : log(relu(x)+0.5) ; else -softplus(-x)
    return (h >= 0.0f) ? logf(h + 0.5f) : -softplusf(-h);
}
__device__ __forceinline__ float logaddexpf_(float p, float q) {
    if (p == -INFINITY) return q;
    if (q == -INFINITY) return p;
    float m = fmaxf(p, q);
    return m + log1pf(expf(-fabsf(p - q)));
}

// ---------------------------------------------------------------------------
// K0: fold the linear chains.
//   Wc[64][1536] = W_in[64][512] @ W_hg[512][1536]
//   Wf[768]      = W_out[768][512] @ W_final[512]
// ---------------------------------------------------------------------------
__global__ __launch_bounds__(256) void k0_fold_weights(
    const float* __restrict__ W_in, const float* __restrict__ W_hg,
    const float* __restrict__ W_out, const float* __restrict__ W_final,
    float* __restrict__ Wc, float* __restrict__ Wf)
{
    int t = blockIdx.x * blockDim.x + threadIdx.x;
    if (t < INDIM * 2 * DI) {                 // 98304 Wc entries
        int i = t / (2 * DI);                 // input row of Wc
        int j = t % (2 * DI);                 // output col
        float acc = 0.0f;
        for (int k = 0; k < HID; ++k)
            acc = fmaf(W_in[i * HID + k], W_hg[k * 2 * DI + j], acc);
        Wc[t] = acc;
    } else if (t < INDIM * 2 * DI + DI) {     // 768 Wf entries
        int d = t - INDIM * 2 * DI;
        float acc = 0.0f;
        for (int k = 0; k < HID; ++k)
            acc = fmaf(W_out[d * HID + k], W_final[k], acc);
        Wf[d] = acc;
    }
}

// ---------------------------------------------------------------------------
// K1: WMMA GEMM  hg = x[65536,64] @ Wc[64,1536]  + gated-log epilogue.
// One wave owns a 16(M)x16(N) tile PAIR: columns [n0,n0+16) ("hidden") and
// [768+n0, 768+n0+16) ("gate"), accumulated with v_wmma_f32_16x16x4_f32 over
// K=64 (16 steps), so the epilogue has h and g for the same channel.
//
// A layout (32-bit A 16x4, ISA 7.12.2): lanes 0-15 carry K=k0,k0+1 in the two
// VGPRs, lanes 16-31 carry K=k0+2,k0+3. B (4x16) mirrored; C/D: VGPR i holds
// row i (lanes 0-15) and row i+8 (lanes 16-31).
// ---------------------------------------------------------------------------
__global__ __launch_bounds__(256) void k1_gemm_hg(
    const float* __restrict__ x, const float* __restrict__ Wc,
    float* __restrict__ LC, float* __restrict__ LV)
{
    const int lane  = threadIdx.x & 31;
    const int wid   = blockIdx.x * 8 + (threadIdx.x >> 5); // 8 waves / block
    const int mt    = wid / (DI / 16);                     // 4096 row tiles
    const int nt    = wid % (DI / 16);                     // 48 col tiles
    const int row0  = mt * 16;
    const int l15   = lane & 15;
    const int khalf = (lane >> 4) ? 2 : 0;                 // upper lanes: K+2
    const int ncol  = nt * 16 + l15;

    const float* arow = x + (size_t)(row0 + l15) * INDIM + khalf;

    v8f acc_h = {};
    v8f acc_g = {};
    #pragma unroll
    for (int k0 = 0; k0 < INDIM; k0 += 4) {
        v2f a;
        a.x = arow[k0];
        a.y = arow[k0 + 1];
        const int kb = k0 + khalf;
        v2f bh, bg;
        bh.x = Wc[(size_t)kb       * (2 * DI) + ncol];
        bh.y = Wc[(size_t)(kb + 1) * (2 * DI) + ncol];
        bg.x = Wc[(size_t)kb       * (2 * DI) + DI + ncol];
        bg.y = Wc[(size_t)(kb + 1) * (2 * DI) + DI + ncol];
        acc_h = __builtin_amdgcn_wmma_f32_16x16x4_f32(
            false, a, false, bh, (short)0, acc_h, false, false);
        acc_g = __builtin_amdgcn_wmma_f32_16x16x4_f32(
            false, a, false, bg, (short)0, acc_g, false, false);
    }

    const int rbase = row0 + ((lane >> 4) ? 8 : 0);
    #pragma unroll
    for (int i = 0; i < 8; ++i) {
        float h = acc_h[i];
        float g = acc_g[i];
        float lc = -softplusf(g);                 // log(1-z)
        float lv = -softplusf(-g) + log_gf(h);    // log(z) + log_g(h)
        size_t o = (size_t)(rbase + i) * DI + ncol;
        LC[o] = lc;
        LV[o] = lv;
    }
}

// ---------------------------------------------------------------------------
// K2: per-chunk scan totals. Thread = (chunk c, channel ch). Streaming
// logsumexp of (lv - a_local) with running cumsum a_local over 256 steps.
// ---------------------------------------------------------------------------
__global__ __launch_bounds__(256) void k2_chunk_totals(
    const float* __restrict__ LC, const float* __restrict__ LV,
    float* __restrict__ Ac, float* __restrict__ Lc)
{
    int t  = blockIdx.x * blockDim.x + threadIdx.x;   // [0, NCHUNK*NCH)
    int ch = t % NCH;
    int c  = t / NCH;
    int b  = ch / DI;
    int d  = ch % DI;
    size_t idx = ((size_t)(b * SEQ + c * CLEN)) * DI + d;

    float a = 0.0f, M = -INFINITY, r = 0.0f;
    for (int j = 0; j < CLEN; ++j, idx += DI) {
        float lc = LC[idx];
        float lv = LV[idx];
        a += lc;
        float tt = lv - a;
        if (tt > M) { r = r * expf(M - tt) + 1.0f; M = tt; }
        else        { r += expf(tt - M); }
    }
    Ac[t] = a;
    Lc[t] = M + logf(r);
}

// ---------------------------------------------------------------------------
// K3: sequential prefix combine over the 16 chunk states per channel.
//   carry into chunk c:  a_pre = sum A_i (i<c),
//                        lse_pre = logaddexp over i<c of (L_i - a_pre_i)
// ---------------------------------------------------------------------------
__global__ __launch_bounds__(256) void k3_chunk_prefix(
    const float* __restrict__ Ac, const float* __restrict__ Lc,
    float* __restrict__ Pa, float* __restrict__ Pl)
{
    int ch = blockIdx.x * blockDim.x + threadIdx.x;   // [0, NCH)
    float a = 0.0f, lse = -INFINITY;
    for (int c = 0; c < NCHUNK; ++c) {
        int o = c * NCH + ch;
        Pa[o] = a;
        Pl[o] = lse;
        lse = logaddexpf_(lse, Lc[o] - a);
        a += Ac[o];
    }
}

// ---------------------------------------------------------------------------
// K4: final scan with carry + deterministic in-wave reduction.
// Wave = (chunk c, batch b, d-group dg); lane owns channel d = dg*32+lane.
// Per step: contrib = Wf[d] * exp(log_h); butterfly-sum over 32 lanes;
// lane 0 writes partial[token][dg]. 6.3 MB instead of a 192 MB out tensor.
// ---------------------------------------------------------------------------
__global__ __launch_bounds__(256) void k4_scan_reduce(
    const float* __restrict__ LC, const float* __restrict__ LV,
    const float* __restrict__ Pa, const float* __restrict__ Pl,
    const float* __restrict__ Wf, float* __restrict__ partial)
{
    const int lane = threadIdx.x & 31;
    const int w    = blockIdx.x * 8 + (threadIdx.x >> 5); // [0, 6144)
    const int c    = w / (BATCH * NDG);
    const int rem  = w % (BATCH * NDG);
    const int b    = rem / NDG;
    const int dg   = rem % NDG;
    const int d    = dg * 32 + lane;
    const int ch   = b * DI + d;

    float a   = Pa[c * NCH + ch];
    float lse = Pl[c * NCH + ch];
    float M   = lse;
    float r   = (lse == -INFINITY) ? 0.0f : 1.0f;
    const float wf = Wf[d];

    size_t idx = ((size_t)(b * SEQ + c * CLEN)) * DI + d;
    for (int j = 0; j < CLEN; ++j, idx += DI) {
        float lc = LC[idx];
        float lv = LV[idx];
        a += lc;
        float tt = lv - a;
        if (tt > M) { r = r * expf(M - tt) + 1.0f; M = tt; }
        else        { r += expf(tt - M); }
        float log_h = a + M + logf(r);
        float val = wf * expf(log_h);
        // deterministic butterfly reduction across the wave (fixed order)
        #pragma unroll
        for (int off = 16; off > 0; off >>= 1)
            val += __shfl_xor(val, off, 32);
        if (lane == 0) {
            int s = c * CLEN + j;
            partial[((size_t)(b * SEQ + s)) * NDG + dg] = val;
        }
    }
}

// ---------------------------------------------------------------------------
// K5: predictions[token] = sum of 24 partials.
// ---------------------------------------------------------------------------
__global__ __launch_bounds__(256) void k5_final_sum(
    const float* __restrict__ partial, float* __restrict__ out)
{
    int t = blockIdx.x * blockDim.x + threadIdx.x;    // [0, NTOK)
    const float* p = partial + (size_t)t * NDG;
    float s = 0.0f;
    #pragma unroll
    for (int g = 0; g < NDG; ++g) s += p[g];
    out[t] = s;
}

// ---------------------------------------------------------------------------
// Launcher. Workspace layout (all fp32, 256B aligned):
//   Wc 384KB | Wf 3KB | LC 192MB | LV 192MB | Ac,Lc,Pa,Pl 4x768KB | partial 6.3MB
// ---------------------------------------------------------------------------
static inline size_t align256(size_t x) { return (x + 255) & ~(size_t)255; }

extern "C" void kernel_launch(void* const* d_in, const int* in_sizes, int n_in,
                              void* d_out, int out_size, void* d_ws, size_t ws_size,
                              hipStream_t stream) {
    const float* x       = (const float*)d_in[0];
    const float* W_in    = (const float*)d_in[1];
    const float* W_hg    = (const float*)d_in[2];
    const float* W_out   = (const float*)d_in[3];
    const float* W_final = (const float*)d_in[4];
    float* out = (float*)d_out;

    char* ws = (char*)d_ws;
    size_t off = 0;
    float* Wc = (float*)(ws + off); off = align256(off + (size_t)INDIM * 2 * DI * 4);
    float* Wf = (float*)(ws + off); off = align256(off + (size_t)DI * 4);
    float* LC = (float*)(ws + off); off = align256(off + (size_t)NTOK * DI * 4);
    float* LV = (float*)(ws + off); off = align256(off + (size_t)NTOK * DI * 4);
    float* Ac = (float*)(ws + off); off = align256(off + (size_t)NCHUNK * NCH * 4);
    float* Lc = (float*)(ws + off); off = align256(off + (size_t)NCHUNK * NCH * 4);
    float* Pa = (float*)(ws + off); off = align256(off + (size_t)NCHUNK * NCH * 4);
    float* Pl = (float*)(ws + off); off = align256(off + (size_t)NCHUNK * NCH * 4);
    float* partial = (float*)(ws + off); off = align256(off + (size_t)NTOK * NDG * 4);
    (void)ws_size; (void)in_sizes; (void)n_in; (void)out_size;

    // K0: fold weights (98304 + 768 outputs)
    {
        int n = INDIM * 2 * DI + DI;
        k0_fold_weights<<<(n + 255) / 256, 256, 0, stream>>>(
            W_in, W_hg, W_out, W_final, Wc, Wf);
    }
    // K1: WMMA GEMM + epilogue. 4096 row-tiles x 48 col-tiles = 196608 waves.
    {
        int nwaves = (NTOK / 16) * (DI / 16);
        k1_gemm_hg<<<nwaves / 8, 256, 0, stream>>>(x, Wc, LC, LV);
    }
    // K2: chunk totals (16 x 12288 threads)
    k2_chunk_totals<<<(NCHUNK * NCH) / 256, 256, 0, stream>>>(LC, LV, Ac, Lc);
    // K3: chunk prefix (12288 threads)
    k3_chunk_prefix<<<NCH / 256, 256, 0, stream>>>(Ac, Lc, Pa, Pl);
    // K4: final scan + wave reduction (6144 waves)
    k4_scan_reduce<<<(NCHUNK * BATCH * NDG) / 8, 256, 0, stream>>>(
        LC, LV, Pa, Pl, Wf, partial);
    // K5: final 24-way sum per token
    k5_final_sum<<<NTOK / 256, 256, 0, stream>>>(partial, out);
}